// SE3BNConvolution_16217796509736
// MI455X (gfx1250) — compile-verified
//
#include <hip/hip_runtime.h>
#include <hip/hip_bf16.h>

// ---------------------------------------------------------------------------
// SE3 BatchNorm + assembled 5x5x5 conv (C_in=C_out=72) as implicit GEMM on
// CDNA5 fp32 WMMA (v_wmma_f32_16x16x4_f32).
//
// Roofline: 340 GFLOP vs ~151 MB minimum HBM traffic -> compute bound
// (AI ~ 2250 FLOP/B). Input (75 MB) + assembled weights (3.2 MB) are
// L2-resident (192 MB), so halo re-reads never touch HBM. fp32 WMMA is the
// precision-exact matrix path for this fp32 reference.
//
// Pipeline (all on `stream`):
//   1) reduce_stats  : per-channel sum / sumsq, 4 deterministic partials each
//   2) finalize_stats: per-channel scale = bn_w * rsqrt(norm2+eps), mean
//   3) assemble_w    : pair-interleaved weights W2[125][40][80][2]
//   4) conv_wmma     : block = (b,x,y); 5 waves, one M-tile each, two 16x16
//                      fp32 accumulators (z 0..15 / 16..31). A frags: one
//                      global b64 (L2/WGP$ hit, prefetched); B frags: one
//                      ds b64 each from a pair-interleaved LDS halo slab.
//                      5 ci-chunks * 125 taps * 4 k-steps * 2 N = 5000
//                      WMMAs per wave over K = 72*125 = 9000.
// ---------------------------------------------------------------------------

typedef float v2f_ __attribute__((ext_vector_type(2)));
typedef float v8f_ __attribute__((ext_vector_type(8)));

#define C_TOT 72
#define C_PAD 80
#define NTAP  125
#define SPAT  (32 * 32 * 32)

// workspace layout (float offsets)
#define WS_SUM   0            // [72][4] partial sums
#define WS_SQ    288          // [72][4] partial sumsq
#define WS_SCALE 576          // [72] per-channel scale
#define WS_MEAN  648          // [72] per-channel mean (nonzero only for c<8)
#define WS_W     704          // [125][40][80][2] pair-interleaved weights
#define WS_W_ELEMS (NTAP * C_PAD * C_PAD)   // 800000 floats (3.2 MB)

// ---------------------------------------------------------------------------
// 1) per-channel statistics: 72 channels x 4 partials, no atomics (determinism)
// ---------------------------------------------------------------------------
__global__ __launch_bounds__(256) void reduce_stats(const float* __restrict__ in,
                                                    float* __restrict__ ws) {
  const int c = blockIdx.x >> 2;   // 0..71
  const int part = blockIdx.x & 3; // 0..3 -> batches {2p, 2p+1}
  float s1 = 0.f, s2 = 0.f;
  for (int b = part * 2; b < part * 2 + 2; ++b) {
    const float* p = in + ((size_t)b * C_TOT + c) * SPAT;
    for (int i = threadIdx.x; i < SPAT; i += 256) {
      float v = p[i];
      s1 += v;
      s2 += v * v;
    }
  }
  __shared__ float r1[256], r2[256];
  r1[threadIdx.x] = s1;
  r2[threadIdx.x] = s2;
  __syncthreads();
  for (int off = 128; off > 0; off >>= 1) {
    if (threadIdx.x < (unsigned)off) {
      r1[threadIdx.x] += r1[threadIdx.x + off];
      r2[threadIdx.x] += r2[threadIdx.x + off];
    }
    __syncthreads();
  }
  if (threadIdx.x == 0) {
    ws[WS_SUM + c * 4 + part] = r1[0];
    ws[WS_SQ + c * 4 + part] = r2[0];
  }
}

// ---------------------------------------------------------------------------
// 2) fold BN into per-channel (scale, mean)
// ---------------------------------------------------------------------------
__global__ __launch_bounds__(128) void finalize_stats(const float* __restrict__ bnw,
                                                      float* __restrict__ ws) {
  const int c = threadIdx.x;
  if (c >= C_TOT) return;
  const float Nel = 8.0f * 32768.0f;
  auto S1 = [&](int ch) {
    return ws[WS_SUM + ch * 4 + 0] + ws[WS_SUM + ch * 4 + 1] +
           ws[WS_SUM + ch * 4 + 2] + ws[WS_SUM + ch * 4 + 3];
  };
  auto S2 = [&](int ch) {
    return ws[WS_SQ + ch * 4 + 0] + ws[WS_SQ + ch * 4 + 1] +
           ws[WS_SQ + ch * 4 + 2] + ws[WS_SQ + ch * 4 + 3];
  };
  float mean = 0.f, norm2, w;
  if (c < 8) {                       // scalar fields: mean-subtract, d=1
    mean = S1(c) / Nel;
    norm2 = S2(c) / Nel - mean * mean;  // E[(x-mu)^2]
    w = bnw[c];
  } else if (c < 32) {               // d=3 fields
    int g = (c - 8) / 3, c0 = 8 + g * 3;
    norm2 = (S2(c0) + S2(c0 + 1) + S2(c0 + 2)) / Nel;
    w = bnw[8 + g];
  } else {                           // d=5 fields
    int g = (c - 32) / 5, c0 = 32 + g * 5;
    norm2 = (S2(c0) + S2(c0 + 1) + S2(c0 + 2) + S2(c0 + 3) + S2(c0 + 4)) / Nel;
    w = bnw[16 + g];
  }
  ws[WS_SCALE + c] = w * rsqrtf(norm2 + 1e-5f);
  ws[WS_MEAN + c] = mean;
}

// ---------------------------------------------------------------------------
// 3) assemble equivariant kernel into pair-interleaved W2[tap][ci/2][co][2]
//    so the GEMM's A fragment (two K rows per lane) is one global b64 load.
// ---------------------------------------------------------------------------
__global__ __launch_bounds__(256) void assemble_w(const float* __restrict__ basis,
                                                  const float* __restrict__ wts,
                                                  float* __restrict__ ws) {
  int idx = blockIdx.x * 256 + threadIdx.x;
  if (idx >= WS_W_ELEMS) return;
  int tap = idx / (C_PAD * C_PAD);
  int rem = idx % (C_PAD * C_PAD);
  int ci = rem / C_PAD;
  int co = rem % C_PAD;
  float v = 0.f;
  if (ci < C_TOT && co < C_TOT) {
    int bo, u, a;
    if (co < 8)       { bo = 0; u = co;            a = 0; }
    else if (co < 32) { bo = 1; u = (co - 8) / 3;  a = (co - 8) % 3; }
    else              { bo = 2; u = (co - 32) / 5; a = (co - 32) % 5; }
    int bi, vv, bb;
    if (ci < 8)       { bi = 0; vv = ci;            bb = 0; }
    else if (ci < 32) { bi = 1; vv = (ci - 8) / 3;  bb = (ci - 8) % 3; }
    else              { bi = 2; vv = (ci - 32) / 5; bb = (ci - 32) % 5; }
    int p = bo * 3 + bi;
#pragma unroll
    for (int n = 0; n < 3; ++n) {
      float wv = wts[((p * 8 + u) * 8 + vv) * 3 + n];                 // [9][8][8][3]
      float bv = basis[(((p * 3 + n) * 5 + a) * 5 + bb) * 125 + tap]; // [9][3][5][5][125]
      v += wv * bv;
    }
  }
  // pair-interleaved: [tap][ci>>1][co][ci&1]
  size_t w2 = (((size_t)tap * (C_PAD / 2) + (ci >> 1)) * C_PAD + co) * 2 + (ci & 1);
  ws[WS_W + w2] = v;
}

// ---------------------------------------------------------------------------
// 4) implicit-GEMM conv with fp32 WMMA.
//    Block = (b,x,y); 160 threads = 5 wave32, wave w owns M-tile w and both
//    N-tiles (two v8f accumulators). LDS slab: [ci_pair(8)][dx(5)][dy(5)]
//    [zz(36)][2] fp32 = 57.6 KB. Staging is row-wise: one thread stages a
//    whole (ci-pair, dx, dy) z-line with b128 global loads + b64 LDS stores.
// ---------------------------------------------------------------------------
__global__ __launch_bounds__(160, 1) void conv_wmma(const float* __restrict__ in,
                                                    const float* __restrict__ ws,
                                                    float* __restrict__ out) {
  __shared__ float slab[8 * 5 * 5 * 36 * 2]; // 14400 floats, 57.6 KB

  const float* scale = ws + WS_SCALE;
  const float* meanc = ws + WS_MEAN;
  const float* W2 = ws + WS_W;

  const int bid = blockIdx.x;
  const int b = bid >> 10;
  const int x = (bid >> 5) & 31;
  const int y = bid & 31;

  const int lane = threadIdx.x & 31;
  const int mtile = threadIdx.x >> 5; // 0..4 -> co rows [16m, 16m+16)
  const int h = lane >> 4;            // half-wave select (K rows 2h, 2h+1)
  const int l16 = lane & 15;
  const int row = mtile * 16 + l16;   // A-matrix M index for this lane

  v8f_ acc0 = {}; // z = 0..15
  v8f_ acc1 = {}; // z = 16..31

  const v2f_ zero2 = {0.f, 0.f};

  for (int cib = 0; cib < C_PAD; cib += 16) {
    // ---- stage BN-normalized halo slab, one (ci-pair, dx, dy) z-line per
    //      iteration: 8x b128 global loads, contiguous b64 LDS stores ----
    for (int rid = threadIdx.x; rid < 200; rid += 160) {
      int kp = rid / 25;          // ci pair 0..7
      int r = rid % 25;           // dx*5+dy
      int dxi = r / 5, dyi = r % 5;
      int gx = x + dxi - 2, gy = y + dyi - 2;
      int c0 = cib + kp * 2, c1 = c0 + 1;
      float* dst = slab + (size_t)(kp * 25 + r) * 72; // 36 pairs per row
      if ((unsigned)gx >= 32u || (unsigned)gy >= 32u || c0 >= C_TOT) {
#pragma unroll
        for (int zz = 0; zz < 36; ++zz) *(v2f_*)(dst + zz * 2) = zero2;
      } else {
        const bool ok1 = (c1 < C_TOT);
        const float mu0 = meanc[c0], sc0 = scale[c0];
        const float mu1 = ok1 ? meanc[c1] : 0.f;
        const float sc1 = ok1 ? scale[c1] : 0.f; // sc1=0 zeroes padded channel
        const float* s0 = in + ((((size_t)b * C_TOT + c0) * 32 + gx) * 32 + gy) * 32;
        const float* s1 = ok1 ? s0 + SPAT : s0;
        *(v2f_*)(dst + 0) = zero2;  // gz = -2
        *(v2f_*)(dst + 2) = zero2;  // gz = -1
        *(v2f_*)(dst + 68) = zero2; // gz = 32
        *(v2f_*)(dst + 70) = zero2; // gz = 33
#pragma unroll
        for (int z4 = 0; z4 < 8; ++z4) {
          float4 a4 = *(const float4*)(s0 + z4 * 4);
          float4 b4 = *(const float4*)(s1 + z4 * 4);
          float* d = dst + (z4 * 4 + 2) * 2;
          v2f_ p0 = {(a4.x - mu0) * sc0, (b4.x - mu1) * sc1};
          v2f_ p1 = {(a4.y - mu0) * sc0, (b4.y - mu1) * sc1};
          v2f_ p2 = {(a4.z - mu0) * sc0, (b4.z - mu1) * sc1};
          v2f_ p3 = {(a4.w - mu0) * sc0, (b4.w - mu1) * sc1};
          *(v2f_*)(d + 0) = p0;
          *(v2f_*)(d + 2) = p1;
          *(v2f_*)(d + 4) = p2;
          *(v2f_*)(d + 6) = p3;
        }
      }
    }
    __syncthreads();

    // ---- GEMM over this 16-channel chunk: 5x5x5 taps x 4 k-steps x 2 N ----
    const size_t wstride = (size_t)(C_PAD / 2) * C_PAD * 2; // floats per tap
    const float* wtap = W2 + ((size_t)(cib >> 1) * C_PAD + row) * 2 +
                        (size_t)h * (C_PAD * 2);
    for (int dx = 0; dx < 5; ++dx) {
      for (int dy = 0; dy < 5; ++dy) {
        const float* sb = slab + ((dx * 5 + dy) * 36 + l16) * 2 + h * 3600;
        for (int dz = 0; dz < 5; ++dz) {
          const float* sp = sb + dz * 2;
          // pull next tap's A-fragment line toward the WGP ahead of use
          __builtin_prefetch(wtap + wstride, 0, 0);
#pragma unroll
          for (int kc = 0; kc < 4; ++kc) {
            // A: W2[tap][cib/2 + kc*2 + h][row][0:2]  (global b64, L2-hit)
            v2f_ af = *(const v2f_*)(wtap + (size_t)kc * (2 * C_PAD * 2));
            // B: slab[kc*4+2h][dx][dy][l16+dz][0:2]     (ds b64)
            v2f_ bf0 = *(const v2f_*)(sp + kc * 7200);
            v2f_ bf1 = *(const v2f_*)(sp + kc * 7200 + 32); // z-tile +16
            acc0 = __builtin_amdgcn_wmma_f32_16x16x4_f32(
                false, af, false, bf0, (short)0, acc0, false, false);
            acc1 = __builtin_amdgcn_wmma_f32_16x16x4_f32(
                false, af, false, bf1, (short)0, acc1, false, false);
          }
          wtap += wstride; // next tap
        }
      }
    }
    __syncthreads();
  }

  // ---- write two 16x16 fp32 tiles: VGPR r holds rows {r, r+8} ----
#pragma unroll
  for (int r = 0; r < 8; ++r) {
    int co = mtile * 16 + r + h * 8;
    if (co < C_TOT) {
      size_t idx = ((((size_t)b * C_TOT + co) * 32 + x) * 32 + y) * 32 + l16;
      out[idx] = acc0[r];
      out[idx + 16] = acc1[r];
    }
  }
}

// ---------------------------------------------------------------------------
extern "C" void kernel_launch(void* const* d_in, const int* in_sizes, int n_in,
                              void* d_out, int out_size, void* d_ws, size_t ws_size,
                              hipStream_t stream) {
  const float* in = (const float*)d_in[0];     // [8,72,32,32,32]
  const float* bnw = (const float*)d_in[1];    // [24]
  const float* basis = (const float*)d_in[2];  // [9,3,5,5,5,5,5]
  const float* wts = (const float*)d_in[3];    // [9,8,8,3]
  float* out = (float*)d_out;                  // [8,72,32,32,32]
  float* ws = (float*)d_ws;                    // needs ~3.3 MB

  (void)in_sizes; (void)n_in; (void)out_size; (void)ws_size;

  reduce_stats<<<C_TOT * 4, 256, 0, stream>>>(in, ws);
  finalize_stats<<<1, 128, 0, stream>>>(bnw, ws);
  assemble_w<<<(WS_W_ELEMS + 255) / 256, 256, 0, stream>>>(basis, wts, ws);
  conv_wmma<<<8 * 32 * 32, 160, 0, stream>>>(in, ws, out);
}